// UnicycleNetwork_89824946029245
// MI455X (gfx1250) — compile-verified
//
#include <hip/hip_runtime.h>

// UnicycleNetwork step, fused for MI455X (gfx1250, wave32).
//
// Inputs (setup_inputs order, all f32):
//  0 u_lin(16,1024) 1 u_ang 2 x 3 z 4 theta 5 s 6 omega
//  7 stiffness(1024,1024) 8 eq_dist(1024,1024,1) 9 ang_coupling(1024,1024)
// 10 eq_ang(1024,1024) 11 lin_damping(1,1024) 12 ang_damping(1,1024)
// 13 mass_vector(1,1024) 14 j_vector(1,1024)
// Output: concat(x_new, z_new, theta_new, s_new, omega_new) = 5*16384 f32.

#define NU   1024
#define NB   16
#define DTC  0.01f
#define CH   128          // j-chunk length staged in LDS
#define CHP  17           // padded LDS row stride (64-bank conflict avoidance)

typedef __attribute__((ext_vector_type(2))) float v2f;
typedef __attribute__((ext_vector_type(8))) float v8f;

// ---------------------------------------------------------------------------
// Kernel 0: batch-independent row sums of the angular coupling matrix:
//   rs_k[i]  = sum_j A[i][j]
//   rs_ke[i] = sum_j A[i][j] * E[i][j]
// One wave32 per row, shuffle-reduced.
// ---------------------------------------------------------------------------
__global__ __launch_bounds__(256) void rowsums_kernel(
    const float* __restrict__ A, const float* __restrict__ E,
    float* __restrict__ rs_k, float* __restrict__ rs_ke) {
  const int wave = threadIdx.x >> 5;
  const int lane = threadIdx.x & 31;
  const int i = (blockIdx.x << 3) + wave;
  const float* Ar = A + (size_t)i * NU;
  const float* Er = E + (size_t)i * NU;
  float sk = 0.f, ske = 0.f;
  for (int j = lane; j < NU; j += 32) {
    const float a = Ar[j];
    sk += a;
    ske = __fmaf_rn(a, Er[j], ske);
  }
  #pragma unroll
  for (int off = 16; off > 0; off >>= 1) {
    sk  += __shfl_down(sk, off, 32);
    ske += __shfl_down(ske, off, 32);
  }
  if (lane == 0) { rs_k[i] = sk; rs_ke[i] = ske; }
}

// ---------------------------------------------------------------------------
// Kernel 1: fused step. Grid = 64 blocks (one per 16-unit tile), 256 thr.
//   Phase B: coup3[b,i] = sum_j A[i][j]*theta[b][j] via V_WMMA_F32_16X16X4_F32,
//            split-K over the 8 waves, reduced with ds_add_f32 LDS atomics.
//   Phase C: all-pairs spring projection; x/z/stiffness/eq staged in LDS and
//            reused across all 16 batches (16x global-traffic reduction).
//   Phase D: integrate + write 5 state outputs.
// ---------------------------------------------------------------------------
__global__ __launch_bounds__(256) void unicycle_kernel(
    const float* __restrict__ u_lin, const float* __restrict__ u_ang,
    const float* __restrict__ x,     const float* __restrict__ z,
    const float* __restrict__ theta, const float* __restrict__ s,
    const float* __restrict__ omega,
    const float* __restrict__ stiff, const float* __restrict__ eqd,
    const float* __restrict__ angc,
    const float* __restrict__ lind,  const float* __restrict__ angd,
    const float* __restrict__ mass,  const float* __restrict__ jvec,
    const float* __restrict__ rs_k,  const float* __restrict__ rs_ke,
    float* __restrict__ out) {
  __shared__ float xj[CH * CHP];
  __shared__ float zj[CH * CHP];
  __shared__ float kf[CH * CHP];
  __shared__ float ef[CH * CHP];
  __shared__ float coup[NB * 16];

  const int tid  = threadIdx.x;
  const int b    = tid >> 4;      // batch 0..15
  const int il   = tid & 15;      // unit within tile
  const int i0   = blockIdx.x << 4;
  const int i    = i0 + il;
  const int lane = tid & 31;
  const int wave = tid >> 5;

  coup[tid] = 0.0f;
  __syncthreads();

  // ---- Phase B: angular GEMM via WMMA (f32 16x16x4), split-K across waves.
  {
    const int half = lane >> 4;   // which K-half of the 16x4 / 4x16 fragment
    const int l16  = lane & 15;
    // A fragment: theta[batch=l16][k], lanes 0-15 hold K{0,1}, 16-31 K{2,3}.
    const float* aRow = theta + (size_t)l16 * NU;
    // B fragment: B[k][n] = ang_coupling[i0+n][k]; same K striping.
    const float* bRow = angc + (size_t)(i0 + l16) * NU;
    v8f acc = {0.f, 0.f, 0.f, 0.f, 0.f, 0.f, 0.f, 0.f};
    const int k_begin = wave * (NU / 8);
    const int k_end   = k_begin + (NU / 8);
    for (int k0 = k_begin; k0 < k_end; k0 += 4) {
      const v2f av = *(const v2f*)(aRow + k0 + 2 * half);
      const v2f bv = *(const v2f*)(bRow + k0 + 2 * half);
      acc = __builtin_amdgcn_wmma_f32_16x16x4_f32(
          /*neg_a=*/false, av, /*neg_b=*/false, bv,
          /*c_mod=*/(short)0, acc, /*reuse_a=*/false, /*reuse_b=*/false);
    }
    // D layout: VGPR r -> M=r (lanes 0-15) / M=r+8 (lanes 16-31), N=l16.
    #pragma unroll
    for (int r = 0; r < 8; ++r) {
      atomicAdd(&coup[(r + 8 * half) * 16 + l16], acc[r]);  // ds_add_f32
    }
  }

  // ---- Phase C: all-pairs spring projection, chunked through LDS.
  const int g   = b * NU + i;
  const float xi = x[g];
  const float zi = z[g];
  const float th = theta[g];
  float sn, cs;
  __sincosf(th, &sn, &cs);
  float proj = 0.0f;

  for (int c0 = 0; c0 < NU; c0 += CH) {
    __syncthreads();  // previous chunk fully consumed (also orders ds_adds)
    // Stage chunk: global reads coalesced, LDS transposed with pad-17 stride.
    #pragma unroll
    for (int idx = tid; idx < CH * NB; idx += 256) {
      const int jrel = idx & (CH - 1);
      const int m    = idx >> 7;             // CH == 128
      xj[jrel * CHP + m] = x[(size_t)m * NU + c0 + jrel];
      zj[jrel * CHP + m] = z[(size_t)m * NU + c0 + jrel];
      kf[jrel * CHP + m] = stiff[(size_t)(i0 + m) * NU + c0 + jrel];
      ef[jrel * CHP + m] = eqd  [(size_t)(i0 + m) * NU + c0 + jrel];
    }
    __syncthreads();
    #pragma unroll 4
    for (int j = 0; j < CH; ++j) {
      const float dx = xi - xj[j * CHP + b];   // LDS broadcast across il
      const float dz = zi - zj[j * CHP + b];
      const float sq = __fmaf_rn(dx, dx, dz * dz);
      const float k  = kf[j * CHP + il];       // conflict-free (pad 17)
      const float e  = ef[j * CHP + il];
      const float inv = (sq > 0.0f) ? __frsqrt_rn(sq) : 0.0f;
      const float dot = __fmaf_rn(dx, cs, dz * sn);  // == 0 when sq == 0
      proj = __fmaf_rn(k * __fmaf_rn(e, inv, -1.0f), dot, proj);
    }
  }
  __syncthreads();  // coup[] and last chunk complete

  // ---- Phase D: combine and integrate.
  const float om = omega[g];
  const float sv = s[g];
  const float coup_ang = rs_ke[i] - th * rs_k[i] + coup[b * 16 + il];
  const float omega_new =
      __fmaf_rn((u_ang[g] + coup_ang - om * angd[i]) * jvec[i], DTC, om);
  const float s_new =
      __fmaf_rn((u_lin[g] + proj - sv * lind[i]) * mass[i], DTC, sv);
  const float theta_new = __fmaf_rn(DTC, omega_new, th);
  float s2, c2;
  __sincosf(theta_new, &s2, &c2);
  const float x_new = __fmaf_rn(c2 * s_new, DTC, xi);
  const float z_new = __fmaf_rn(s2 * s_new, DTC, zi);

  out[0 * NB * NU + g] = x_new;
  out[1 * NB * NU + g] = z_new;
  out[2 * NB * NU + g] = theta_new;
  out[3 * NB * NU + g] = s_new;
  out[4 * NB * NU + g] = omega_new;
}

extern "C" void kernel_launch(void* const* d_in, const int* in_sizes, int n_in,
                              void* d_out, int out_size, void* d_ws, size_t ws_size,
                              hipStream_t stream) {
  (void)in_sizes; (void)n_in; (void)out_size; (void)ws_size;
  const float* u_lin = (const float*)d_in[0];
  const float* u_ang = (const float*)d_in[1];
  const float* x     = (const float*)d_in[2];
  const float* z     = (const float*)d_in[3];
  const float* theta = (const float*)d_in[4];
  const float* s     = (const float*)d_in[5];
  const float* omega = (const float*)d_in[6];
  const float* stiff = (const float*)d_in[7];
  const float* eqd   = (const float*)d_in[8];
  const float* angc  = (const float*)d_in[9];
  const float* eqang = (const float*)d_in[10];
  const float* lind  = (const float*)d_in[11];
  const float* angd  = (const float*)d_in[12];
  const float* mass  = (const float*)d_in[13];
  const float* jvec  = (const float*)d_in[14];
  float* out = (float*)d_out;

  float* rs_k  = (float*)d_ws;         // 1024 floats
  float* rs_ke = rs_k + NU;            // 1024 floats

  rowsums_kernel<<<NU / 8, 256, 0, stream>>>(angc, eqang, rs_k, rs_ke);
  unicycle_kernel<<<NU / 16, 256, 0, stream>>>(
      u_lin, u_ang, x, z, theta, s, omega, stiff, eqd, angc,
      lind, angd, mass, jvec, rs_k, rs_ke, out);
}